// GATNet_22471268892725
// MI455X (gfx1250) — compile-verified
//
#include <hip/hip_runtime.h>
#include <hip/hip_bf16.h>

#define NNODES 50000
#define NEDGES 800000
#define F_IN   128
#define HID    32
#define HEADS  8
#define HC     256           // HEADS*HID
#define NCLS   40
#define NEG_SLOPE 0.2f

typedef float v2f __attribute__((ext_vector_type(2)));
typedef float v8f __attribute__((ext_vector_type(8)));

// ---------------- monotonic float<->uint encoding for atomic max ----------
__device__ __forceinline__ unsigned f2ord(float f) {
    unsigned u = __float_as_uint(f);
    return (u & 0x80000000u) ? ~u : (u | 0x80000000u);
}
__device__ __forceinline__ float ord2f(unsigned u) {
    return (u & 0x80000000u) ? __uint_as_float(u & 0x7FFFFFFFu)
                             : __uint_as_float(~u);
}

// ---------------- WMMA fp32 GEMM: C[M x Nc] = A[M x K] @ B[K x Nc] (+bias) ----
// One wave32 computes NT adjacent 16x16 output tiles (shared A fragment) with
// V_WMMA_F32_16X16X4_F32.
// A fragment (ISA 7.12.2, 32-bit A 16x4): lane l (l<16), half = lane>=16:
//   a = { A[row0+l][k+2*half], A[row0+l][k+2*half+1] }
// B fragment (4x16): b = { B[k+2*half][col], B[k+2*half+1][col] }, col = col0+l.
// C/D (32-bit 16x16): VGPR r -> C[row0 + r + 8*half][col0 + l].
template <int NT, bool RAGGED>
__global__ void gat_wmma_gemm_f32(const float* __restrict__ A,
                                  const float* __restrict__ B,
                                  float* __restrict__ C,
                                  const float* __restrict__ bias,
                                  int M, int K, int Nc) {
    int wave   = (blockIdx.x * blockDim.x + threadIdx.x) >> 5;
    int lane   = threadIdx.x & 31;
    int groups = ((Nc + 15) >> 4) / NT + ((((Nc + 15) >> 4) % NT) ? 1 : 0);
    int tm = wave / groups;
    int tg = wave - tm * groups;
    if (tm * 16 >= M) return;               // wave-uniform exit
    int row0 = tm << 4;
    int col0 = (tg * NT) << 4;
    int half = lane >> 4;                   // 0 or 1
    int l    = lane & 15;

    // column index / mask (only the ragged path ever masks)
    int   cbase = col0 + l;
    int   cclmp = cbase;
    float cmask = 1.0f;
    if (RAGGED) {
        cclmp = (cbase < Nc) ? cbase : (Nc - 1);
        cmask = (cbase < Nc) ? 1.0f : 0.0f;
    }

    const float* ap = A + (size_t)(row0 + l) * K + 2 * half;
    const float* bp = B + (size_t)(2 * half) * Nc + (RAGGED ? cclmp : cbase);
    const size_t bstep = (size_t)4 * Nc;

    v8f acc[NT];
    #pragma unroll
    for (int t = 0; t < NT; ++t) acc[t] = (v8f){};

    for (int k = 0; k < K; k += 4) {
        // issue all loads of this k-step into distinct registers first so the
        // scheduler can overlap them and use partial loadcnt waits
        v2f a;
        a.x = ap[0];
        a.y = ap[1];
        ap += 4;
        float b0[NT], b1[NT];
        #pragma unroll
        for (int t = 0; t < NT; ++t) {
            b0[t] = bp[t * 16];
            b1[t] = bp[(size_t)Nc + t * 16];
            if (RAGGED) { b0[t] *= cmask; b1[t] *= cmask; }
        }
        bp += bstep;
        #pragma unroll
        for (int t = 0; t < NT; ++t) {
            v2f b;
            b.x = b0[t];
            b.y = b1[t];
            acc[t] = __builtin_amdgcn_wmma_f32_16x16x4_f32(
                /*neg_a=*/false, a, /*neg_b=*/false, b,
                /*c_mod=*/(short)0, acc[t], /*reuse_a=*/false, /*reuse_b=*/false);
        }
    }

    #pragma unroll
    for (int t = 0; t < NT; ++t) {
        int c = cbase + t * 16;
        if (RAGGED && c >= Nc) continue;
        float bv = bias ? bias[c] : 0.0f;
        float* cp = C + (size_t)(row0 + 8 * half) * Nc + c;
        #pragma unroll
        for (int r = 0; r < 8; ++r)
            cp[(size_t)r * Nc] = acc[t][r] + bv;
    }
}

// ---------------- attention source/dest scores: a_s[n,h] = <h[n,h,:], att_s[h,:]> ----
__global__ void gat_att_scores(const float* __restrict__ H,
                               const float* __restrict__ att_s,
                               const float* __restrict__ att_d,
                               float* __restrict__ as_out,
                               float* __restrict__ ad_out) {
    int i = blockIdx.x * blockDim.x + threadIdx.x;   // over N*HEADS
    if (i >= NNODES * HEADS) return;
    int n = i >> 3;
    int h = i & 7;
    const float* hp = H + (size_t)n * HC + h * HID;
    const float* sp = att_s + h * HID;
    const float* dp = att_d + h * HID;
    float accs = 0.0f, accd = 0.0f;
    #pragma unroll
    for (int cix = 0; cix < HID; ++cix) {
        float v = hp[cix];
        accs += v * sp[cix];
        accd += v * dp[cix];
    }
    as_out[i] = accs;
    ad_out[i] = accd;
}

// ---------------- edge pass 1: leaky-relu logits + segment max (ordered-uint) ----
__global__ void gat_edge_logits(const int* __restrict__ src,
                                const int* __restrict__ dst,
                                const float* __restrict__ as_,
                                const float* __restrict__ ad_,
                                float* __restrict__ e_buf,
                                unsigned* __restrict__ mbuf,
                                int etot) {
    int i = blockIdx.x * blockDim.x + threadIdx.x;   // over etot*HEADS
    if (i >= etot * HEADS) return;
    int e = i >> 3;
    int h = i & 7;
    int s = (e < NEDGES) ? src[e] : (e - NEDGES);    // self-loops appended
    int d = (e < NEDGES) ? dst[e] : (e - NEDGES);
    float v = as_[s * HEADS + h] + ad_[d * HEADS + h];
    v = (v > 0.0f) ? v : NEG_SLOPE * v;
    e_buf[i] = v;
    atomicMax(mbuf + d * HEADS + h, f2ord(v));
}

// ---------------- edge pass 2: exp(e - max) + segment sum ----------------
__global__ void gat_edge_exp(const int* __restrict__ src,
                             const int* __restrict__ dst,
                             const unsigned* __restrict__ mbuf,
                             float* __restrict__ e_buf,
                             float* __restrict__ denom,
                             int etot) {
    int i = blockIdx.x * blockDim.x + threadIdx.x;
    if (i >= etot * HEADS) return;
    int e = i >> 3;
    int h = i & 7;
    int d = (e < NEDGES) ? dst[e] : (e - NEDGES);
    float mv = ord2f(mbuf[d * HEADS + h]);
    float ex = __expf(e_buf[i] - mv);
    e_buf[i] = ex;
    atomicAdd(denom + d * HEADS + h, ex);
}

// ---------------- edge pass 3: out[dst] += h[src] * alpha (wave per edge) ----
__global__ void gat_edge_aggregate(const int* __restrict__ src,
                                   const int* __restrict__ dst,
                                   const float* __restrict__ H,
                                   const float* __restrict__ e_buf,
                                   const float* __restrict__ denom,
                                   float* __restrict__ out,
                                   int etot) {
    int lane = threadIdx.x & 31;
    int e = (blockIdx.x * blockDim.x + threadIdx.x) >> 5;
    if (e >= etot) return;
    int s = (e < NEDGES) ? src[e] : (e - NEDGES);
    int d = (e < NEDGES) ? dst[e] : (e - NEDGES);
    const float* hp = H + (size_t)s * HC;
    float*       op = out + (size_t)d * HC;
    #pragma unroll
    for (int j = 0; j < HEADS; ++j) {
        int cix = lane + 32 * j;                     // head index == j
        float alpha = e_buf[(size_t)e * HEADS + j]
                      / (denom[d * HEADS + j] + 1e-16f);
        atomicAdd(op + cix, hp[cix] * alpha);
    }
}

// ---------------- bias (+ optional ELU) elementwise ----------------------
__global__ void gat_bias_act(float* __restrict__ out,
                             const float* __restrict__ bias,
                             int total, int do_elu) {
    int i = blockIdx.x * blockDim.x + threadIdx.x;
    if (i >= total) return;
    float v = out[i] + bias[i & (HC - 1)];
    if (do_elu) v = (v > 0.0f) ? v : (__expf(v) - 1.0f);
    out[i] = v;
}

// ---------------- host-side orchestration --------------------------------
static inline void launch_gemm_aligned(const float* A, const float* B, float* C,
                                       const float* bias, int M, int K, int Nc,
                                       hipStream_t stream) {
    constexpr int NT = 4;                            // Nc must be multiple of 64
    int groups = (Nc / 16) / NT;
    int waves  = ((M + 15) / 16) * groups;
    int blocks = (waves + 7) / 8;                    // 8 waves / 256-thread block
    gat_wmma_gemm_f32<NT, false><<<blocks, 256, 0, stream>>>(A, B, C, bias, M, K, Nc);
}

static inline void launch_gemm_ragged(const float* A, const float* B, float* C,
                                      const float* bias, int M, int K, int Nc,
                                      hipStream_t stream) {
    int groups = (Nc + 15) / 16;
    int waves  = ((M + 15) / 16) * groups;
    int blocks = (waves + 7) / 8;
    gat_wmma_gemm_f32<1, true><<<blocks, 256, 0, stream>>>(A, B, C, bias, M, K, Nc);
}

extern "C" void kernel_launch(void* const* d_in, const int* in_sizes, int n_in,
                              void* d_out, int out_size, void* d_ws, size_t ws_size,
                              hipStream_t stream) {
    const float* x   = (const float*)d_in[0];
    const int*   ei  = (const int*)d_in[1];          // [2, E]
    const float* w1  = (const float*)d_in[2];
    const float* as1 = (const float*)d_in[3];
    const float* ad1 = (const float*)d_in[4];
    const float* b1  = (const float*)d_in[5];
    const float* w2  = (const float*)d_in[6];
    const float* as2 = (const float*)d_in[7];
    const float* ad2 = (const float*)d_in[8];
    const float* b2  = (const float*)d_in[9];
    const float* wc  = (const float*)d_in[10];
    const float* bc  = (const float*)d_in[11];
    const int* src = ei;
    const int* dst = ei + NEDGES;

    const int etot = NEDGES + NNODES;                // with self-loops
    // scratch layout (floats)
    float* ws      = (float*)d_ws;
    float* h_buf   = ws;                                         // N*HC
    float* out_buf = h_buf + (size_t)NNODES * HC;                // N*HC
    float* a_s     = out_buf + (size_t)NNODES * HC;              // N*H
    float* a_d     = a_s + (size_t)NNODES * HEADS;               // N*H
    unsigned* mbuf = (unsigned*)(a_d + (size_t)NNODES * HEADS);  // N*H
    float* denom   = (float*)(mbuf + (size_t)NNODES * HEADS);    // N*H
    float* e_buf   = denom + (size_t)NNODES * HEADS;             // etot*H

    const int nh      = NNODES * HEADS;
    const int eh      = etot * HEADS;
    const int nhBlk   = (nh + 255) / 256;
    const int ehBlk   = (eh + 255) / 256;
    const int aggBlk  = (etot * 32 + 255) / 256;
    const int ewBlk   = (NNODES * HC + 255) / 256;

    // ---------------- layer 1 ----------------
    launch_gemm_aligned(x, w1, h_buf, nullptr, NNODES, F_IN, HC, stream);
    gat_att_scores<<<nhBlk, 256, 0, stream>>>(h_buf, as1, ad1, a_s, a_d);
    hipMemsetAsync(mbuf,    0, (size_t)nh * 4, stream);   // ordered-uint 0 == -inf
    hipMemsetAsync(denom,   0, (size_t)nh * 4, stream);
    hipMemsetAsync(out_buf, 0, (size_t)NNODES * HC * 4, stream);
    gat_edge_logits<<<ehBlk, 256, 0, stream>>>(src, dst, a_s, a_d, e_buf, mbuf, etot);
    gat_edge_exp<<<ehBlk, 256, 0, stream>>>(src, dst, mbuf, e_buf, denom, etot);
    gat_edge_aggregate<<<aggBlk, 256, 0, stream>>>(src, dst, h_buf, e_buf, denom,
                                                   out_buf, etot);
    gat_bias_act<<<ewBlk, 256, 0, stream>>>(out_buf, b1, NNODES * HC, /*elu=*/1);

    // ---------------- layer 2 ----------------
    launch_gemm_aligned(out_buf, w2, h_buf, nullptr, NNODES, HC, HC, stream);
    gat_att_scores<<<nhBlk, 256, 0, stream>>>(h_buf, as2, ad2, a_s, a_d);
    hipMemsetAsync(mbuf,    0, (size_t)nh * 4, stream);
    hipMemsetAsync(denom,   0, (size_t)nh * 4, stream);
    hipMemsetAsync(out_buf, 0, (size_t)NNODES * HC * 4, stream);
    gat_edge_logits<<<ehBlk, 256, 0, stream>>>(src, dst, a_s, a_d, e_buf, mbuf, etot);
    gat_edge_exp<<<ehBlk, 256, 0, stream>>>(src, dst, mbuf, e_buf, denom, etot);
    gat_edge_aggregate<<<aggBlk, 256, 0, stream>>>(src, dst, h_buf, e_buf, denom,
                                                   out_buf, etot);
    gat_bias_act<<<ewBlk, 256, 0, stream>>>(out_buf, b2, NNODES * HC, /*elu=*/0);

    // ---------------- classifier ----------------
    launch_gemm_ragged(out_buf, wc, (float*)d_out, bc, NNODES, HC, NCLS, stream);
}